// Rotation_69114613728558
// MI455X (gfx1250) — compile-verified
//
#include <hip/hip_runtime.h>

// Problem constants (from reference): x is N x D, D=128, P = D*(D-1)/2 = 8128 rotations.
#define DD 128
#define NPAIRS 8128
#define NCHUNK 16
#define CHUNK_LEN (NPAIRS / NCHUNK)   // 508
#define MAT_ELEMS (DD * DD)           // 16384 floats
#define MAT_BYTES (MAT_ELEMS * 4)     // 65536 bytes
#define APPLY_GRID 256                // persistent blocks for the streaming GEMM

// LDS pad strides (words). Chosen so WMMA fragment gathers are bank-conflict free:
//  A-frag rows: stride 132 -> bank = (4*row + k) % 64, distinct for 16 rows, halves disjoint.
//  B-frag rows: stride 136 -> 2*stride % 64 == 16, halves land on disjoint bank ranges.
#define XSTRIDE 132
#define YSTRIDE 136

typedef __attribute__((ext_vector_type(2))) float v2f;
typedef __attribute__((ext_vector_type(8))) float v8f;
typedef __attribute__((ext_vector_type(4))) int v4i;

#ifndef __has_builtin
#define __has_builtin(x) 0
#endif
#if __has_builtin(__builtin_amdgcn_global_load_async_to_lds_b128)
#define HAVE_ASYNC_COPY 1
#else
#define HAVE_ASYNC_COPY 0
#endif

// ---------------------------------------------------------------------------
// K1: parallel cos/sin of all thetas.
// ---------------------------------------------------------------------------
__global__ void Rot_cs_kernel(const float* __restrict__ theta,
                              float2* __restrict__ cs) {
  int k = blockIdx.x * blockDim.x + threadIdx.x;
  if (k < NPAIRS) {
    float t = theta[k];
    cs[k] = make_float2(cosf(t), sinf(t));
  }
}

// ---------------------------------------------------------------------------
// K2: each block builds the product of CHUNK_LEN Givens rotations (in order)
// starting from identity. Thread t owns row t of the partial product E:
//   E := E * A_k  touches only columns (i,j):  E[t][i], E[t][j].
// Rows are independent -> no barriers in the sequential loop.
// LDS row stride 129 (odd) -> bank = (t + col) % 64, conflict-free per wave.
// ---------------------------------------------------------------------------
__global__ void __launch_bounds__(128)
Rot_chunk_build_kernel(const int2* __restrict__ pairs,
                       const float2* __restrict__ cs,
                       float* __restrict__ chunks) {
  __shared__ float E[DD * 129];
  __shared__ int2 sp[CHUNK_LEN];
  __shared__ float2 scs[CHUNK_LEN];
  const int t = threadIdx.x;                 // 0..127, one row per thread
  const int k0 = blockIdx.x * CHUNK_LEN;

  float* row = &E[t * 129];
  for (int c = 0; c < DD; ++c) row[c] = (c == t) ? 1.0f : 0.0f;
  for (int q = t; q < CHUNK_LEN; q += 128) {
    sp[q] = pairs[k0 + q];
    scs[q] = cs[k0 + q];
  }
  __syncthreads();

  for (int q = 0; q < CHUNK_LEN; ++q) {
    const int2 ij = sp[q];
    const float2 c_s = scs[q];
    const float ei = row[ij.x];
    const float ej = row[ij.y];
    row[ij.x] = c_s.x * ei - c_s.y * ej;
    row[ij.y] = c_s.y * ei + c_s.x * ej;
  }
  __syncthreads();

  // Fully coalesced write-out: 4096 float4s via 128 threads (32 iters),
  // each wave covers one dense 512B row segment per iteration.
  float* out = chunks + blockIdx.x * MAT_ELEMS;
  for (int it = 0; it < 32; ++it) {
    int f = it * 128 + t;           // float4 index 0..4095
    int r = f >> 5;
    int c4 = f & 31;
    const float* s = &E[r * 129 + c4 * 4];
    float4 v = make_float4(s[0], s[1], s[2], s[3]);
    *((float4*)(out + r * DD) + c4) = v;
  }
}

// ---------------------------------------------------------------------------
// Shared WMMA GEMM core (f32 16x16x4): wave computes a 16-row x 128-col stripe.
// ---------------------------------------------------------------------------

// A-matrix 16x4 f32 fragment (ISA 7.12.2): lane L: row = L%16,
// VGPR0 = K = (L/16)*2, VGPR1 = K+1.
__device__ __forceinline__ v2f load_a_frag(const float* lds, int row0, int kbase) {
  const int lane = threadIdx.x & 31;
  const float* p = lds + (row0 + (lane & 15)) * XSTRIDE + kbase + ((lane >> 4) << 1);
  v2f a;
  a.x = p[0];
  a.y = p[1];
  return a;
}

// B-matrix 4x16 f32 fragment (row striped across lanes per VGPR):
// lane L: col = L%16; VGPR v holds K = (L/16)*2 + v.
__device__ __forceinline__ v2f load_b_frag(const float* lds, int kbase, int col0) {
  const int lane = threadIdx.x & 31;
  const int col = col0 + (lane & 15);
  const int k = kbase + ((lane >> 4) << 1);
  v2f b;
  b.x = lds[k * YSTRIDE + col];
  b.y = lds[(k + 1) * YSTRIDE + col];
  return b;
}

// Copy a dense 128x128 f32 tile (leading dim ldg) into padded LDS, 256 threads.
template <int STRIDE>
__device__ __forceinline__ void stage_tile(const float* __restrict__ g, int ldg,
                                           float* lds) {
  for (int it = 0; it < 16; ++it) {
    int f = it * 256 + threadIdx.x;   // float4 index 0..4095
    int r = f >> 5;
    int c4 = f & 31;
    float4 v = *((const float4*)(g + r * ldg) + c4);
    float* d = lds + r * STRIDE + c4 * 4;
    d[0] = v.x; d[1] = v.y; d[2] = v.z; d[3] = v.w;
  }
}

// Stage a 128x128 x-tile into padded LDS. Async path: each wave issues 16
// GLOBAL_LOAD_ASYNC_TO_LDS_B128 covering its 16 rows; every instruction spans
// one dense 512B row segment (lanes 0..31 -> consecutive float4s of one row).
__device__ __forceinline__ void load_x_tile(const float* __restrict__ g,
                                            float* lbuf) {
#if HAVE_ASYNC_COPY
  const int wave = threadIdx.x >> 5;
  const int lane = threadIdx.x & 31;
#pragma unroll
  for (int it = 0; it < 16; ++it) {
    int f = it * 32 + lane;                  // float4 index within wave's rows
    int r = wave * 16 + (f >> 5);
    int c4 = f & 31;
    const float* src = g + r * DD + c4 * 4;
    float* dst = lbuf + r * XSTRIDE + c4 * 4;
    __builtin_amdgcn_global_load_async_to_lds_b128(
        (__attribute__((address_space(1))) v4i*)src,
        (__attribute__((address_space(3))) v4i*)dst,
        /*imm_offset=*/0, /*cpol=*/0);
  }
#else
  stage_tile<XSTRIDE>(g, DD, lbuf);
#endif
}

__device__ __forceinline__ void wait_x_tile() {
#if HAVE_ASYNC_COPY
#if __has_builtin(__builtin_amdgcn_s_wait_asynccnt)
  __builtin_amdgcn_s_wait_asynccnt(0);
#else
  asm volatile("s_wait_asynccnt 0" ::: "memory");
#endif
#endif
}

// 8 waves/block; wave w computes rows [16w,16w+16) x all 128 cols:
// 8 accumulator tiles, K = 128 in 32 steps of V_WMMA_F32_16X16X4_F32.
__device__ __forceinline__ void wave_gemm_stripe(const float* Xlds, const float* Ylds,
                                                 v8f acc[8]) {
  const int row0 = (threadIdx.x >> 5) * 16;
#pragma unroll
  for (int t = 0; t < 8; ++t) {
    v8f z = {};
    acc[t] = z;
  }
  for (int kt = 0; kt < 32; ++kt) {
    v2f a = load_a_frag(Xlds, row0, kt * 4);
#pragma unroll
    for (int t = 0; t < 8; ++t) {
      v2f b = load_b_frag(Ylds, kt * 4, t * 16);
      acc[t] = __builtin_amdgcn_wmma_f32_16x16x4_f32(
          /*neg_a=*/false, a, /*neg_b=*/false, b,
          /*c_mod=*/(short)0, acc[t], /*reuse_a=*/false, /*reuse_b=*/false);
    }
  }
}

// C/D 16x16 f32 layout: lane L -> col = L%16, VGPR v -> row = v + (L/16)*8.
__device__ __forceinline__ void store_stripe(float* g, int ldg, const v8f acc[8]) {
  const int lane = threadIdx.x & 31;
  const int row0 = (threadIdx.x >> 5) * 16 + ((lane >> 4) << 3);
  const int col = lane & 15;
#pragma unroll
  for (int t = 0; t < 8; ++t) {
#pragma unroll
    for (int v = 0; v < 8; ++v) {
      g[(row0 + v) * ldg + t * 16 + col] = acc[t][v];
    }
  }
}

// ---------------------------------------------------------------------------
// K3 (tree combine): block b computes dst[b] = src[2b] * src[2b+1] (128^3 GEMM).
// ---------------------------------------------------------------------------
__global__ void __launch_bounds__(256)
Rot_pair_product_kernel(const float* __restrict__ src, float* __restrict__ dst) {
  __shared__ float Xlds[DD * XSTRIDE];
  __shared__ float Ylds[DD * YSTRIDE];
  const float* X = src + (size_t)(2 * blockIdx.x) * MAT_ELEMS;
  const float* Y = src + (size_t)(2 * blockIdx.x + 1) * MAT_ELEMS;
  stage_tile<XSTRIDE>(X, DD, Xlds);
  stage_tile<YSTRIDE>(Y, DD, Ylds);
  __syncthreads();
  v8f acc[8];
  wave_gemm_stripe(Xlds, Ylds, acc);
  store_stripe(dst + (size_t)blockIdx.x * MAT_ELEMS, DD, acc);
}

// ---------------------------------------------------------------------------
// K4: out = x * M. Persistent blocks: M staged once, x tiles double-buffered
// through LDS with async global->LDS copies overlapping the WMMA loop.
// Memory-bound: 64MB read + 64MB write -> ~5.5us floor at 23.3 TB/s.
// LDS: M (69,632B) + 2 x-buffers (2*67,584B) = 204,800B -> 1 block/WGP, 8 waves.
// ---------------------------------------------------------------------------
__global__ void __launch_bounds__(256)
Rot_apply_kernel(const float* __restrict__ x, const float* __restrict__ M,
                 float* __restrict__ out, int ntiles) {
  __shared__ float Mlds[DD * YSTRIDE];
  __shared__ float Xlds[2][DD * XSTRIDE];

  stage_tile<YSTRIDE>(M, DD, Mlds);          // once per persistent block

  int tile = blockIdx.x;
  const int stride = gridDim.x;
  if (tile < ntiles) {
    load_x_tile(x + (size_t)tile * DD * DD, Xlds[0]);
  }
  wait_x_tile();
  __syncthreads();                           // M + first x tile visible

  int buf = 0;
  while (tile < ntiles) {
    const int next = tile + stride;
    if (next < ntiles) {
      load_x_tile(x + (size_t)next * DD * DD, Xlds[buf ^ 1]);  // prefetch
    }
    v8f acc[8];
    wave_gemm_stripe(Xlds[buf], Mlds, acc);
    store_stripe(out + (size_t)tile * DD * DD, DD, acc);
    wait_x_tile();                           // own async copies done
    __syncthreads();                         // everyone's copies visible
    buf ^= 1;
    tile = next;
  }
}

// ---------------------------------------------------------------------------
// Launch. Inputs: d_in[0]=x (N*128 f32), d_in[1]=theta (8128 f32),
// d_in[2]=pairs (8128*2 i32). Workspace use: 25 * 64KB = 1.6 MB.
// ---------------------------------------------------------------------------
extern "C" void kernel_launch(void* const* d_in, const int* in_sizes, int n_in,
                              void* d_out, int out_size, void* d_ws, size_t ws_size,
                              hipStream_t stream) {
  const float* x = (const float*)d_in[0];
  const float* theta = (const float*)d_in[1];
  const int2* pairs = (const int2*)d_in[2];
  float* out = (float*)d_out;

  char* ws = (char*)d_ws;
  float2* cs = (float2*)ws;                                   // 64 KB slot
  float* bufA = (float*)(ws + 65536);                         // 16 matrices
  float* bufB = (float*)(ws + 65536 + (size_t)NCHUNK * MAT_BYTES);  // 8 matrices

  const int N = in_sizes[0] / DD;  // 131072
  const int ntiles = N / DD;       // 1024

  // 1) cos/sin table (parallel).
  Rot_cs_kernel<<<(NPAIRS + 255) / 256, 256, 0, stream>>>(theta, cs);
  // 2) 16 chunk products of 508 rotations each -> bufA[0..15].
  Rot_chunk_build_kernel<<<NCHUNK, 128, 0, stream>>>(pairs, cs, bufA);
  // 3) Order-preserving tree reduction: 16 -> 8 -> 4 -> 2 -> 1 (final M = bufA[0]).
  Rot_pair_product_kernel<<<8, 256, 0, stream>>>(bufA, bufB);
  Rot_pair_product_kernel<<<4, 256, 0, stream>>>(bufB, bufA);
  Rot_pair_product_kernel<<<2, 256, 0, stream>>>(bufA, bufB);
  Rot_pair_product_kernel<<<1, 256, 0, stream>>>(bufB, bufA);
  // 4) out = x * M, persistent double-buffered streaming GEMM with f32 WMMA.
  Rot_apply_kernel<<<APPLY_GRID, 256, 0, stream>>>(x, bufA, out, ntiles);
}